// EncodeChrom_87797721464881
// MI455X (gfx1250) — compile-verified
//
#include <hip/hip_runtime.h>
#include <hip/hip_bf16.h>

#define N_NODES   2000
#define NUM_GRAPHS 64
#define N_TOTAL   (N_NODES * NUM_GRAPHS)   // 128000
#define N_EDGES   2000000
#define F_IN      9
#define F_OUT     64
#define HID       128
#define NEG_SLOPE 0.2f
#define EN        (N_EDGES + N_TOTAL)      // edges + self loops

typedef __attribute__((ext_vector_type(2))) float v2f;
typedef __attribute__((ext_vector_type(8))) float v8f;

// -------- monotonic float<->uint mapping for atomicMax on floats --------
__device__ __forceinline__ unsigned f2ord(float f) {
    unsigned u = __float_as_uint(f);
    return (u & 0x80000000u) ? ~u : (u | 0x80000000u);
}
__device__ __forceinline__ float ord2f(unsigned o) {
    return __uint_as_float((o & 0x80000000u) ? (o ^ 0x80000000u) : ~o);
}

// out[n,f] = b1[f] + b2[f]  (bias of both convs folded into the accumulator)
__global__ void k_init_out(const float* __restrict__ b1, const float* __restrict__ b2,
                           float* __restrict__ out) {
    int idx = blockIdx.x * blockDim.x + threadIdx.x;
    if (idx < N_TOTAL * F_OUT) out[idx] = b1[idx & 63] + b2[idx & 63];
}

__global__ void k_init_seg(unsigned* __restrict__ emax, float* __restrict__ esum) {
    int i = blockIdx.x * blockDim.x + threadIdx.x;
    if (i < N_TOTAL) { emax[i] = 0u; esum[i] = 0.0f; }
}

// xw = x @ W ; alpha_s = xw @ a_src ; alpha_d = xw @ a_dst   (one thread per node)
__global__ void k_transform(const float* __restrict__ x, const float* __restrict__ W,
                            const float* __restrict__ a_src, const float* __restrict__ a_dst,
                            float* __restrict__ xw, float* __restrict__ as_, float* __restrict__ ad_) {
    __shared__ float sW[F_IN * F_OUT];
    __shared__ float sas[F_OUT], sad[F_OUT];
    for (int i = threadIdx.x; i < F_IN * F_OUT; i += blockDim.x) sW[i] = W[i];
    for (int i = threadIdx.x; i < F_OUT; i += blockDim.x) { sas[i] = a_src[i]; sad[i] = a_dst[i]; }
    __syncthreads();
    int n = blockIdx.x * blockDim.x + threadIdx.x;
    if (n >= N_TOTAL) return;
    float xr[F_IN];
#pragma unroll
    for (int k = 0; k < F_IN; ++k) xr[k] = x[n * F_IN + k];
    float as = 0.f, ad = 0.f;
    for (int f = 0; f < F_OUT; ++f) {
        float acc = 0.f;
#pragma unroll
        for (int k = 0; k < F_IN; ++k) acc = fmaf(xr[k], sW[k * F_OUT + f], acc);
        xw[n * F_OUT + f] = acc;
        as = fmaf(acc, sas[f], as);
        ad = fmaf(acc, sad[f], ad);
    }
    as_[n] = as; ad_[n] = ad;
}

// pass 1: e = leaky(alpha_s[src] + alpha_d[dst]); per-dst running max
__global__ void k_edge_pass1(const int* __restrict__ ei, const float* __restrict__ as_,
                             const float* __restrict__ ad_, float* __restrict__ ebuf,
                             unsigned* __restrict__ emax) {
    int idx = blockIdx.x * blockDim.x + threadIdx.x;
    if (idx >= EN) return;
    int src, dst;
    if (idx < N_EDGES) { src = ei[idx]; dst = ei[N_EDGES + idx]; }
    else               { src = dst = idx - N_EDGES; }
    float e = as_[src] + ad_[dst];
    e = (e >= 0.f) ? e : NEG_SLOPE * e;
    ebuf[idx] = e;
    atomicMax(&emax[dst], f2ord(e));
}

// pass 2: e_exp = exp(e - max[dst]); per-dst sum
__global__ void k_edge_pass2(const int* __restrict__ ei, float* __restrict__ ebuf,
                             const unsigned* __restrict__ emax, float* __restrict__ esum) {
    int idx = blockIdx.x * blockDim.x + threadIdx.x;
    if (idx >= EN) return;
    int dst = (idx < N_EDGES) ? ei[N_EDGES + idx] : (idx - N_EDGES);
    float ex = __expf(ebuf[idx] - ord2f(emax[dst]));
    ebuf[idx] = ex;
    atomicAdd(&esum[dst], ex);
}

// pass 3: out[dst,f] += (e_exp / sum[dst]) * xw[src,f]   (thread per edge-feature)
__global__ void k_edge_pass3(const int* __restrict__ ei, const float* __restrict__ ebuf,
                             const float* __restrict__ esum, const float* __restrict__ xw,
                             float* __restrict__ out) {
    int t = blockIdx.x * blockDim.x + threadIdx.x;   // EN*64 = 136,192,000 < 2^31
    if (t >= EN * F_OUT) return;
    int e = t >> 6, f = t & 63;
    int src, dst;
    if (e < N_EDGES) { src = ei[e]; dst = ei[N_EDGES + e]; }
    else             { src = dst = e - N_EDGES; }
    float alpha = ebuf[e] / esum[dst];
    atomicAdd(&out[dst * F_OUT + f], alpha * xw[src * F_OUT + f]);
}

// MLP via V_WMMA_F32_16X16X4_F32: one wave per 16-node tile.
// h[16,128] = feat[16,64] @ lin1_w[64,128] kept in 8 f32 accumulators,
// then lin2 ([128]->1) fused as per-lane FMAs + shfl_xor reduction.
__global__ void k_mlp_wmma(const float* __restrict__ feat, const float* __restrict__ lin1_w,
                           const float* __restrict__ lin1_b, const float* __restrict__ lin2_w,
                           const float* __restrict__ lin2_b, float* __restrict__ logits) {
    int wave = (blockIdx.x * blockDim.x + threadIdx.x) >> 5;
    int lane = threadIdx.x & 31;
    if (wave >= N_TOTAL / 16) return;          // exact grid: never taken, wave-uniform
    int m0   = wave * 16;
    int half = lane >> 4;                      // K/M half split per ISA layout
    int mrow = lane & 15;

    v8f zero = {};
    v8f acc[8];
#pragma unroll
    for (int i = 0; i < 8; ++i) acc[i] = zero;

    // A 16x4 f32 layout: lanes 0-15 hold K=k0,k0+1 ; lanes 16-31 hold K=k0+2,k0+3
    const float* arow = feat + (m0 + mrow) * F_OUT + 2 * half;
    for (int ks = 0; ks < 16; ++ks) {          // K = 64 in steps of 4
        int k0 = ks * 4;
        v2f a;
        a.x = arow[k0];
        a.y = arow[k0 + 1];
#pragma unroll
        for (int nt = 0; nt < 8; ++nt) {       // 8 N-tiles of 16 -> HID=128
            const float* bcol = lin1_w + (k0 + 2 * half) * HID + nt * 16 + mrow;
            v2f b;
            b.x = bcol[0];
            b.y = bcol[HID];
            acc[nt] = __builtin_amdgcn_wmma_f32_16x16x4_f32(
                false, a, false, b, (short)0, acc[nt], false, false);
        }
    }

    // Fused lin2: logit[row] = sum_feat (h + lin1_b[feat]) * lin2_w[feat] + lin2_b
    float l2b = lin2_b[0];
#pragma unroll
    for (int j = 0; j < 8; ++j) {              // D VGPR j holds row j (lanes<16) / row 8+j
        float p = 0.f;
#pragma unroll
        for (int nt = 0; nt < 8; ++nt) {
            int fi = nt * 16 + mrow;
            float h = acc[nt][j] + lin1_b[fi];
            p = fmaf(h, lin2_w[fi], p);
        }
#pragma unroll
        for (int off = 1; off < 16; off <<= 1) // reduce within each 16-lane half
            p += __shfl_xor(p, off, 32);
        if (mrow == 0) logits[m0 + half * 8 + j] = p + l2b;
    }
}

// per-graph softmax over 2000 nodes
__global__ void k_softmax(const float* __restrict__ logits, float* __restrict__ out) {
    int g = blockIdx.x;
    const float* l = logits + g * N_NODES;
    __shared__ float sred[8];
    int tid = threadIdx.x;

    float m = -3.4e38f;
    for (int i = tid; i < N_NODES; i += 256) m = fmaxf(m, l[i]);
    for (int off = 16; off; off >>= 1) m = fmaxf(m, __shfl_xor(m, off, 32));
    if ((tid & 31) == 0) sred[tid >> 5] = m;
    __syncthreads();
    if (tid == 0) {
        float t = sred[0];
        for (int w = 1; w < 8; ++w) t = fmaxf(t, sred[w]);
        sred[0] = t;
    }
    __syncthreads();
    float bm = sred[0];
    __syncthreads();

    float s = 0.f;
    for (int i = tid; i < N_NODES; i += 256) s += __expf(l[i] - bm);
    for (int off = 16; off; off >>= 1) s += __shfl_xor(s, off, 32);
    if ((tid & 31) == 0) sred[tid >> 5] = s;
    __syncthreads();
    if (tid == 0) {
        float t = 0.f;
        for (int w = 0; w < 8; ++w) t += sred[w];
        sred[0] = t;
    }
    __syncthreads();
    float inv = 1.f / sred[0];
    for (int i = tid; i < N_NODES; i += 256) out[g * N_NODES + i] = __expf(l[i] - bm) * inv;
}

extern "C" void kernel_launch(void* const* d_in, const int* in_sizes, int n_in,
                              void* d_out, int out_size, void* d_ws, size_t ws_size,
                              hipStream_t stream) {
    (void)in_sizes; (void)n_in; (void)out_size; (void)ws_size;

    const float* x1     = (const float*)d_in[0];
    const float* x2     = (const float*)d_in[1];
    const int*   ei1    = (const int*)d_in[2];
    const int*   ei2    = (const int*)d_in[3];
    const float* W1     = (const float*)d_in[4];
    const float* as1    = (const float*)d_in[5];
    const float* ad1    = (const float*)d_in[6];
    const float* b1     = (const float*)d_in[7];
    const float* W2     = (const float*)d_in[8];
    const float* as2    = (const float*)d_in[9];
    const float* ad2    = (const float*)d_in[10];
    const float* b2     = (const float*)d_in[11];
    const float* lin1w  = (const float*)d_in[12];
    const float* lin1b  = (const float*)d_in[13];
    const float* lin2w  = (const float*)d_in[14];
    const float* lin2b  = (const float*)d_in[15];

    // workspace layout (floats), conv buffers reused between conv1/conv2
    float*    xw     = (float*)d_ws;                 // 8,192,000
    float*    alphas = xw + (size_t)N_TOTAL * F_OUT; // 128,000
    float*    alphad = alphas + N_TOTAL;             // 128,000
    unsigned* emax   = (unsigned*)(alphad + N_TOTAL);// 128,000
    float*    esum   = (float*)emax + N_TOTAL;       // 128,000
    float*    ebuf   = esum + N_TOTAL;               // 2,128,000
    float*    outf   = ebuf + EN;                    // 8,192,000
    float*    logits = outf + (size_t)N_TOTAL * F_OUT;

    const int B = 256;
    unsigned gN   = (N_TOTAL + B - 1) / B;
    unsigned gNF  = (N_TOTAL * F_OUT + B - 1) / B;
    unsigned gE   = (EN + B - 1) / B;
    unsigned gEF  = (unsigned)(((long long)EN * F_OUT + B - 1) / B);

    k_init_out<<<gNF, B, 0, stream>>>(b1, b2, outf);

    // ---- conv 1 ----
    k_transform<<<gN, B, 0, stream>>>(x1, W1, as1, ad1, xw, alphas, alphad);
    k_init_seg<<<gN, B, 0, stream>>>(emax, esum);
    k_edge_pass1<<<gE, B, 0, stream>>>(ei1, alphas, alphad, ebuf, emax);
    k_edge_pass2<<<gE, B, 0, stream>>>(ei1, ebuf, emax, esum);
    k_edge_pass3<<<gEF, B, 0, stream>>>(ei1, ebuf, esum, xw, outf);

    // ---- conv 2 (buffers reused) ----
    k_transform<<<gN, B, 0, stream>>>(x2, W2, as2, ad2, xw, alphas, alphad);
    k_init_seg<<<gN, B, 0, stream>>>(emax, esum);
    k_edge_pass1<<<gE, B, 0, stream>>>(ei2, alphas, alphad, ebuf, emax);
    k_edge_pass2<<<gE, B, 0, stream>>>(ei2, ebuf, emax, esum);
    k_edge_pass3<<<gEF, B, 0, stream>>>(ei2, ebuf, esum, xw, outf);

    // ---- fused MLP (f32 WMMA) + per-graph softmax ----
    k_mlp_wmma<<<(N_TOTAL / 16) / 8, B, 0, stream>>>(outf, lin1w, lin1b, lin2w, lin2b, logits);
    k_softmax<<<NUM_GRAPHS, B, 0, stream>>>(logits, (float*)d_out);
}